// VMoE_1967095022280
// MI455X (gfx1250) — compile-verified
//
#include <hip/hip_runtime.h>
#include <hip/hip_bf16.h>

typedef float v2f __attribute__((ext_vector_type(2)));
typedef float v8f __attribute__((ext_vector_type(8)));

#define EMSIZE   768
#define NHEADS   12
#define NHID     3072
#define NLAYERS  8
#define NEXPERT  8
#define NCLASSES 1000
#define BATCHN   256
#define SEQLEN   65
#define NTOK     (BATCHN * SEQLEN)   // 16640
#define CAPC     2600                // int(1.25*16640/8)

// ---------------------------------------------------------------------------
// Shared WMMA epilogue: C/D 16x16 f32 layout -> VGPR r: lanes0-15 M=r, lanes16-31 M=8+r
// ---------------------------------------------------------------------------
__device__ __forceinline__ void store_tile_f32(const v8f c, float* __restrict__ C,
                                               int M, int N, int ldc, int gm0, int gn,
                                               const float* __restrict__ bias, int act)
{
  if (gn < N) {
    float bv = bias ? bias[gn] : 0.0f;
    #pragma unroll
    for (int r = 0; r < 8; ++r) {
      int gm = gm0 + r;
      if (gm < M) {
        float v = c[r] + bv;
        if (act) v = fmaxf(v, 0.0f);
        C[(long)gm * ldc + gn] = v;
      }
    }
  }
}

// ===========================================================================
// Generic (synchronous) fp32 WMMA GEMM: handles any N / unaligned ldb.
// Block tile 64x64, 4 waves, each 32x32. Used only for the decoder GEMM.
// ===========================================================================
#define BM 64
#define BN 64
#define BK 16
#define AS_LD 20   // padded A row stride (mult of 4)
#define BS_LD 68   // padded B row stride (mult of 4)

__launch_bounds__(128)
__global__ void gemm_wmma_f32(const float* __restrict__ A, const float* __restrict__ B,
                              const float* __restrict__ bias, float* __restrict__ Cmat,
                              int M, int N, int K, int lda, int ldb, int ldc,
                              const int* __restrict__ a_idx, int act)
{
  __shared__ float As[BM * AS_LD];
  __shared__ float Bs[BK * BS_LD];

  const int tid  = threadIdx.x;
  const int lane = tid & 31;
  const int wave = tid >> 5;
  const int m_tile = blockIdx.y * BM;
  const int n_tile = blockIdx.x * BN;
  const int wm = (wave >> 1) << 5;
  const int wn = (wave & 1) << 5;
  const int half = lane >> 4;
  const int l16  = lane & 15;
  const bool bvec = ((ldb & 3) == 0);

  v8f c00 = {0.f,0.f,0.f,0.f,0.f,0.f,0.f,0.f};
  v8f c01 = c00, c10 = c00, c11 = c00;

  for (int k0 = 0; k0 < K; k0 += BK) {
    #pragma unroll
    for (int s = tid; s < 256; s += 128) {
      int row = s >> 2;
      int c4  = (s & 3) << 2;
      float4 v = {0.f, 0.f, 0.f, 0.f};
      int gr = m_tile + row;
      if (gr < M) {
        int ar = a_idx ? a_idx[gr] : gr;
        if (ar >= 0)
          v = *reinterpret_cast<const float4*>(A + (long)ar * lda + (k0 + c4));
      }
      *reinterpret_cast<float4*>(&As[row * AS_LD + c4]) = v;
    }
    #pragma unroll
    for (int s = tid; s < 256; s += 128) {
      int row = s >> 4;
      int c16 = (s & 15) << 2;
      int gn  = n_tile + c16;
      const float* bp = B + (long)(k0 + row) * ldb + gn;
      float4 v = {0.f, 0.f, 0.f, 0.f};
      if (bvec && (gn + 3 < N)) {
        v = *reinterpret_cast<const float4*>(bp);
      } else {
        if (gn + 0 < N) v.x = bp[0];
        if (gn + 1 < N) v.y = bp[1];
        if (gn + 2 < N) v.z = bp[2];
        if (gn + 3 < N) v.w = bp[3];
      }
      *reinterpret_cast<float4*>(&Bs[row * BS_LD + c16]) = v;
    }
    __syncthreads();

    #pragma unroll
    for (int kk = 0; kk < BK; kk += 4) {
      const int r0 = kk + half * 2;
      v2f a0 = *reinterpret_cast<const v2f*>(&As[(wm      + l16) * AS_LD + r0]);
      v2f a1 = *reinterpret_cast<const v2f*>(&As[(wm + 16 + l16) * AS_LD + r0]);
      v2f b0, b1;
      b0.x = Bs[ r0      * BS_LD + wn      + l16];
      b0.y = Bs[(r0 + 1) * BS_LD + wn      + l16];
      b1.x = Bs[ r0      * BS_LD + wn + 16 + l16];
      b1.y = Bs[(r0 + 1) * BS_LD + wn + 16 + l16];
      c00 = __builtin_amdgcn_wmma_f32_16x16x4_f32(false, a0, false, b0, (short)0, c00, false, false);
      c01 = __builtin_amdgcn_wmma_f32_16x16x4_f32(false, a0, false, b1, (short)0, c01, false, false);
      c10 = __builtin_amdgcn_wmma_f32_16x16x4_f32(false, a1, false, b0, (short)0, c10, false, false);
      c11 = __builtin_amdgcn_wmma_f32_16x16x4_f32(false, a1, false, b1, (short)0, c11, false, false);
    }
    __syncthreads();
  }

  const int gn0 = n_tile + wn + l16;
  store_tile_f32(c00, Cmat, M, N, ldc, m_tile + wm      + half * 8, gn0,      bias, act);
  store_tile_f32(c01, Cmat, M, N, ldc, m_tile + wm      + half * 8, gn0 + 16, bias, act);
  store_tile_f32(c10, Cmat, M, N, ldc, m_tile + wm + 16 + half * 8, gn0,      bias, act);
  store_tile_f32(c11, Cmat, M, N, ldc, m_tile + wm + 16 + half * 8, gn0 + 16, bias, act);
}

// ===========================================================================
// Fast-path GEMM: double-buffered LDS filled by GLOBAL_LOAD_ASYNC_TO_LDS_B128
// (ASYNCcnt), prefetching K-tile t+1 while WMMAs consume tile t.
// Requires N % 128 == 0, lda/ldb mult of 4, K mult of 16. M-edge / gather
// rows <0 are zero-filled via predicated ds stores (async honors EXEC).
// Block tile 64x128, 4 waves (2x2), each wave 32x64 (8 accumulators).
// ===========================================================================
#define FBM 64
#define FBN 128
#define FBK 16
#define FB_LD 132   // padded B row stride (mult of 4)

__launch_bounds__(128)
__global__ void gemm_wmma_f32_async(const float* __restrict__ A, const float* __restrict__ B,
                                    const float* __restrict__ bias, float* __restrict__ Cmat,
                                    int M, int N, int K, int lda, int ldb, int ldc,
                                    const int* __restrict__ a_idx, int act)
{
  __shared__ float As[2][FBM * AS_LD];
  __shared__ float Bs[2][FBK * FB_LD];

  const int tid  = threadIdx.x;
  const int lane = tid & 31;
  const int wave = tid >> 5;
  const int m_tile = blockIdx.y * FBM;
  const int n_tile = blockIdx.x * FBN;
  const int wm = (wave >> 1) << 5;   // 0 or 32
  const int wn = (wave & 1) << 6;    // 0 or 64
  const int half = lane >> 4;
  const int l16  = lane & 15;

  v8f acc[2][4];
  #pragma unroll
  for (int i = 0; i < 2; ++i)
    #pragma unroll
    for (int j = 0; j < 4; ++j)
      acc[i][j] = (v8f){0.f,0.f,0.f,0.f,0.f,0.f,0.f,0.f};

  // Issue async DMA (global -> LDS) for one 64x16 A tile + 16x128 B tile.
  auto stage = [&](int kt, int buf) {
    const int k0 = kt * FBK;
    #pragma unroll
    for (int s = tid; s < 256; s += 128) {          // A: 256 float4 slots
      int row = s >> 2, c4 = (s & 3) << 2;
      int gr  = m_tile + row;
      unsigned la = (unsigned)(uintptr_t)&As[buf][row * AS_LD + c4];
      const float* gp = nullptr;
      bool valid = false;
      if (gr < M) {
        int ar = a_idx ? a_idx[gr] : gr;
        if (ar >= 0) { valid = true; gp = A + (long)ar * lda + (k0 + c4); }
      }
      if (valid) {
        asm volatile("global_load_async_to_lds_b128 %0, %1, off"
                     :: "v"(la), "v"(gp) : "memory");
      } else {
        float4 z = {0.f, 0.f, 0.f, 0.f};
        *reinterpret_cast<float4*>(&As[buf][row * AS_LD + c4]) = z;
      }
    }
    #pragma unroll
    for (int s = tid; s < 512; s += 128) {          // B: 512 float4 slots
      int row = s >> 5, c4 = (s & 31) << 2;
      unsigned la = (unsigned)(uintptr_t)&Bs[buf][row * FB_LD + c4];
      const float* gp = B + (long)(k0 + row) * ldb + (n_tile + c4);
      asm volatile("global_load_async_to_lds_b128 %0, %1, off"
                   :: "v"(la), "v"(gp) : "memory");
    }
  };

  const int ntiles = K / FBK;
  stage(0, 0);

  for (int kt = 0; kt < ntiles; ++kt) {
    const int cur = kt & 1;
    asm volatile("s_wait_asynccnt 0x0" ::: "memory");   // tile kt landed in LDS
    __syncthreads();                                    // all waves see it; prev buf free
    if (kt + 1 < ntiles) stage(kt + 1, cur ^ 1);        // prefetch overlaps compute

    #pragma unroll
    for (int kk = 0; kk < FBK; kk += 4) {
      const int r0 = kk + half * 2;                     // ISA 16x4 half-wave K offset
      v2f a0 = *reinterpret_cast<const v2f*>(&As[cur][(wm      + l16) * AS_LD + r0]);
      v2f a1 = *reinterpret_cast<const v2f*>(&As[cur][(wm + 16 + l16) * AS_LD + r0]);
      #pragma unroll
      for (int j = 0; j < 4; ++j) {
        v2f b;
        b.x = Bs[cur][ r0      * FB_LD + wn + j * 16 + l16];
        b.y = Bs[cur][(r0 + 1) * FB_LD + wn + j * 16 + l16];
        acc[0][j] = __builtin_amdgcn_wmma_f32_16x16x4_f32(false, a0, false, b, (short)0, acc[0][j], false, false);
        acc[1][j] = __builtin_amdgcn_wmma_f32_16x16x4_f32(false, a1, false, b, (short)0, acc[1][j], false, false);
      }
    }
  }

  #pragma unroll
  for (int i = 0; i < 2; ++i) {
    const int gm0 = m_tile + wm + i * 16 + half * 8;
    #pragma unroll
    for (int j = 0; j < 4; ++j) {
      const int gn = n_tile + wn + j * 16 + l16;       // always < N (N % 128 == 0)
      float bv = bias ? bias[gn] : 0.0f;
      #pragma unroll
      for (int r = 0; r < 8; ++r) {
        int gm = gm0 + r;
        if (gm < M) {
          float v = acc[i][j][r] + bv;
          if (act) v = fmaxf(v, 0.0f);
          Cmat[(long)gm * ldc + gn] = v;
        }
      }
    }
  }
}

// ---------------------------------------------------------------------------
// Patch rearrange: x[B,3,32,32] -> P[B*64, 48]
// ---------------------------------------------------------------------------
__global__ void patch_rearrange_kernel(const float* __restrict__ x, float* __restrict__ P)
{
  long id = (long)blockIdx.x * blockDim.x + threadIdx.x;
  const long total = (long)BATCHN * 64 * 48;
  if (id >= total) return;
  int f = id % 48;
  int t = (id / 48) % 64;
  int b = id / (48 * 64);
  int c  = f >> 4;
  int a  = (f >> 2) & 3;
  int bb = f & 3;
  int i  = t >> 3;
  int j  = t & 7;
  P[id] = x[(((long)(b * 3 + c) * 32) + i * 4 + a) * 32 + j * 4 + bb];
}

// h[b,t,:] = (t==0 ? cls : patch_out[b*64+t-1]) + pos[t]
__global__ void assemble_kernel(const float* __restrict__ pe, const float* __restrict__ cls,
                                const float* __restrict__ pos, float* __restrict__ h)
{
  long id = (long)blockIdx.x * blockDim.x + threadIdx.x;
  const long total = (long)NTOK * EMSIZE;
  if (id >= total) return;
  int d = id % EMSIZE;
  int s = id / EMSIZE;
  int t = s % SEQLEN;
  int b = s / SEQLEN;
  float v = (t == 0) ? cls[d] : pe[((long)(b * 64 + t - 1)) * EMSIZE + d];
  h[id] = v + pos[(long)t * EMSIZE + d];
}

// ---------------------------------------------------------------------------
// Attention: one block per (batch, head). K/V/scores in LDS (~50 KB).
// ---------------------------------------------------------------------------
__launch_bounds__(128)
__global__ void attn_kernel(const float* __restrict__ qkv, float* __restrict__ out)
{
  const int bh = blockIdx.x;
  const int b  = bh / NHEADS, hh = bh % NHEADS;
  __shared__ float Ks[SEQLEN * 64];
  __shared__ float Vs[SEQLEN * 64];
  __shared__ float Sc[SEQLEN * 66];
  const int tid = threadIdx.x;
  const long base = (long)b * SEQLEN * 2304 + hh * 64;

  for (int idx = tid; idx < SEQLEN * 64; idx += 128) {
    int row = idx >> 6, d = idx & 63;
    long o = base + (long)row * 2304 + d;
    Ks[idx] = qkv[o + 768];
    Vs[idx] = qkv[o + 1536];
  }
  __syncthreads();

  const int t = tid;
  if (t < SEQLEN) {
    float* sc = &Sc[t * 66];
    for (int j = 0; j < SEQLEN; ++j) sc[j] = 0.f;
    const long qo = base + (long)t * 2304;
    for (int c = 0; c < 64; c += 16) {
      float q[16];
      #pragma unroll
      for (int d = 0; d < 16; ++d) q[d] = qkv[qo + c + d];
      for (int j = 0; j < SEQLEN; ++j) {
        float acc = sc[j];
        const float* kr = &Ks[j * 64 + c];   // broadcast across lanes
        #pragma unroll
        for (int d = 0; d < 16; ++d) acc += q[d] * kr[d];
        sc[j] = acc;
      }
    }
    const float scale = 0.125f;   // 1/sqrt(64)
    float m = -3.4e38f;
    for (int j = 0; j < SEQLEN; ++j) { sc[j] *= scale; m = fmaxf(m, sc[j]); }
    float sum = 0.f;
    for (int j = 0; j < SEQLEN; ++j) { float e = expf(sc[j] - m); sc[j] = e; sum += e; }
    float inv = 1.f / sum;
    for (int j = 0; j < SEQLEN; ++j) sc[j] *= inv;
    float* orow = out + ((long)(b * SEQLEN + t) * EMSIZE + hh * 64);
    for (int d = 0; d < 64; ++d) {
      float acc = 0.f;
      for (int j = 0; j < SEQLEN; ++j) acc += sc[j] * Vs[j * 64 + d];
      orow[d] = acc;
    }
  }
}

// ---------------------------------------------------------------------------
// Residual + LayerNorm: out[row] = LN(x[row] + f[row]) * g + b   (in-place ok)
// ---------------------------------------------------------------------------
__global__ void add_ln_kernel(const float* __restrict__ x, const float* __restrict__ f,
                              const float* __restrict__ g, const float* __restrict__ bta,
                              float* __restrict__ out)
{
  const int row = blockIdx.x, tid = threadIdx.x;
  __shared__ float buf[EMSIZE];
  __shared__ float red[256];
  const float* xr = x + (long)row * EMSIZE;
  const float* fr = f + (long)row * EMSIZE;
  float s = 0.f;
  for (int d = tid; d < EMSIZE; d += 256) { float v = xr[d] + fr[d]; buf[d] = v; s += v; }
  red[tid] = s; __syncthreads();
  for (int off = 128; off > 0; off >>= 1) { if (tid < off) red[tid] += red[tid + off]; __syncthreads(); }
  const float mean = red[0] * (1.f / EMSIZE);
  __syncthreads();
  float s2 = 0.f;
  for (int d = tid; d < EMSIZE; d += 256) { float v = buf[d] - mean; s2 += v * v; }
  red[tid] = s2; __syncthreads();
  for (int off = 128; off > 0; off >>= 1) { if (tid < off) red[tid] += red[tid + off]; __syncthreads(); }
  const float rstd = 1.f / sqrtf(red[0] * (1.f / EMSIZE) + 1e-5f);
  float* orow = out + (long)row * EMSIZE;
  for (int d = tid; d < EMSIZE; d += 256)
    orow[d] = (buf[d] - mean) * rstd * g[d] + bta[d];
}

// ---------------------------------------------------------------------------
// MoE gating
// ---------------------------------------------------------------------------
__global__ void gate_logits_kernel(const float* __restrict__ h, const float* __restrict__ gw,
                                   float* __restrict__ logits, int S)
{
  const int wave = threadIdx.x >> 5, lane = threadIdx.x & 31;
  const int s = blockIdx.x * 8 + wave;
  if (s >= S) return;
  float acc[NEXPERT] = {0.f,0.f,0.f,0.f,0.f,0.f,0.f,0.f};
  const float* xr = h + (long)s * EMSIZE;
  for (int d = lane; d < EMSIZE; d += 32) {
    float xv = xr[d];
    const float* gr = gw + d * NEXPERT;
    #pragma unroll
    for (int e = 0; e < NEXPERT; ++e) acc[e] += xv * gr[e];
  }
  #pragma unroll
  for (int e = 0; e < NEXPERT; ++e)
    for (int off = 16; off > 0; off >>= 1)
      acc[e] += __shfl_xor(acc[e], off, 32);
  if (lane == 0) {
    #pragma unroll
    for (int e = 0; e < NEXPERT; ++e) logits[(long)s * NEXPERT + e] = acc[e];
  }
}

__global__ void gate_top2_kernel(const float* __restrict__ logits, int* __restrict__ e1,
                                 int* __restrict__ e2, float* __restrict__ gate1,
                                 float* __restrict__ gate2, int S)
{
  const int s = blockIdx.x * blockDim.x + threadIdx.x;
  if (s >= S) return;
  float l[NEXPERT];
  #pragma unroll
  for (int e = 0; e < NEXPERT; ++e) l[e] = logits[(long)s * NEXPERT + e];
  int a = 0; float mv = l[0];
  #pragma unroll
  for (int e = 1; e < NEXPERT; ++e) if (l[e] > mv) { mv = l[e]; a = e; }   // first max
  int b = -1; float mv2 = -3.4e38f;
  #pragma unroll
  for (int e = 0; e < NEXPERT; ++e) if (e != a && l[e] > mv2) { mv2 = l[e]; b = e; }
  float sum = 0.f;
  #pragma unroll
  for (int e = 0; e < NEXPERT; ++e) sum += expf(l[e] - mv);
  e1[s] = a; e2[s] = b;
  gate1[s] = 1.f / sum;               // exp(0)/sum
  gate2[s] = expf(mv2 - mv) / sum;
}

__global__ void slot_init_kernel(int* __restrict__ slot, int n)
{
  int i = blockIdx.x * blockDim.x + threadIdx.x;
  if (i < n) slot[i] = -1;
}

// Sequential per-expert scan reproducing cumsum-based positions.
__global__ void moe_pos_kernel(const int* __restrict__ e1, const int* __restrict__ e2,
                               int* __restrict__ p1, int* __restrict__ p2, int S)
{
  const int e = threadIdx.x;
  if (e >= NEXPERT) return;
  int c = 0;
  for (int s = 0; s < S; ++s) if (e1[s] == e) { p1[s] = c; ++c; }
  int c2 = 0;
  for (int s = 0; s < S; ++s) if (e2[s] == e) { p2[s] = c + c2; ++c2; }  // offset by m1.sum()
}

__global__ void moe_finalize_kernel(const float* __restrict__ gate1, const float* __restrict__ gate2,
                                    const int* __restrict__ e1, const int* __restrict__ e2,
                                    const int* __restrict__ p1, const int* __restrict__ p2,
                                    float* __restrict__ g1, float* __restrict__ g2,
                                    int* __restrict__ p1c, int* __restrict__ p2c,
                                    int* __restrict__ slot, int S, int C)
{
  const int s = blockIdx.x * blockDim.x + threadIdx.x;
  if (s >= S) return;
  const int q1 = p1[s], q2 = p2[s];
  const float k1 = (q1 < C) ? 1.f : 0.f;
  const float k2 = (q2 < C) ? 1.f : 0.f;
  const float a = gate1[s] * k1, b = gate2[s] * k2;
  const float denom = fmaxf(a + b, 1e-9f);
  g1[s] = a / denom; g2[s] = b / denom;
  p1c[s] = min(q1, C - 1); p2c[s] = min(q2, C - 1);
  if (q1 < C) slot[e1[s] * C + q1] = s;
  if (q2 < C) slot[e2[s] * C + q2] = s;
}

__global__ void moe_combine_kernel(const float* __restrict__ oe, const int* __restrict__ e1,
                                   const int* __restrict__ e2, const int* __restrict__ p1c,
                                   const int* __restrict__ p2c, const float* __restrict__ g1,
                                   const float* __restrict__ g2, float* __restrict__ out,
                                   int S, int C)
{
  long id = (long)blockIdx.x * blockDim.x + threadIdx.x;
  const long total = (long)S * (EMSIZE / 4);
  if (id >= total) return;
  const int s = id / (EMSIZE / 4), r = id % (EMSIZE / 4);
  const float4 v1 = reinterpret_cast<const float4*>(oe + (long)(e1[s] * C + p1c[s]) * EMSIZE)[r];
  const float4 v2 = reinterpret_cast<const float4*>(oe + (long)(e2[s] * C + p2c[s]) * EMSIZE)[r];
  const float a = g1[s], b = g2[s];
  float4 v;
  v.x = a * v1.x + b * v2.x;
  v.y = a * v1.y + b * v2.y;
  v.z = a * v1.z + b * v2.z;
  v.w = a * v1.w + b * v2.w;
  reinterpret_cast<float4*>(out)[id] = v;
}

// ---------------------------------------------------------------------------
// NLL loss
// ---------------------------------------------------------------------------
__global__ void loss_per_b_kernel(const float* __restrict__ logits, const int* __restrict__ y,
                                  float* __restrict__ lossb)
{
  const int b = blockIdx.x, tid = threadIdx.x;
  __shared__ float red[256];
  const float* lr = logits + (long)b * NCLASSES;
  float m = -3.4e38f;
  for (int j = tid; j < NCLASSES; j += 256) m = fmaxf(m, lr[j]);
  red[tid] = m; __syncthreads();
  for (int off = 128; off > 0; off >>= 1) { if (tid < off) red[tid] = fmaxf(red[tid], red[tid + off]); __syncthreads(); }
  m = red[0]; __syncthreads();
  float s = 0.f;
  for (int j = tid; j < NCLASSES; j += 256) s += expf(lr[j] - m);
  red[tid] = s; __syncthreads();
  for (int off = 128; off > 0; off >>= 1) { if (tid < off) red[tid] += red[tid + off]; __syncthreads(); }
  if (tid == 0) lossb[b] = -(lr[y[b]] - m - logf(red[0]));
}

__global__ void loss_reduce_kernel(const float* __restrict__ lossb, float* __restrict__ out)
{
  __shared__ float red[256];
  const int tid = threadIdx.x;
  red[tid] = lossb[tid]; __syncthreads();
  for (int off = 128; off > 0; off >>= 1) { if (tid < off) red[tid] += red[tid + off]; __syncthreads(); }
  if (tid == 0) out[0] = red[0];
}

// ---------------------------------------------------------------------------
// Host launcher
// ---------------------------------------------------------------------------
static inline void launch_gemm(const float* A, const float* B, const float* bias, float* C,
                               int M, int N, int K, int lda, int ldb, int ldc,
                               const int* a_idx, int act, hipStream_t stream)
{
  if ((N % FBN) == 0 && (ldb & 3) == 0 && (lda & 3) == 0 && (K % FBK) == 0) {
    dim3 grid(N / FBN, (M + FBM - 1) / FBM);
    gemm_wmma_f32_async<<<grid, 128, 0, stream>>>(A, B, bias, C, M, N, K, lda, ldb, ldc, a_idx, act);
  } else {
    dim3 grid((N + BN - 1) / BN, (M + BM - 1) / BM);
    gemm_wmma_f32<<<grid, 128, 0, stream>>>(A, B, bias, C, M, N, K, lda, ldb, ldc, a_idx, act);
  }
}

extern "C" void kernel_launch(void* const* d_in, const int* in_sizes, int n_in,
                              void* d_out, int out_size, void* d_ws, size_t ws_size,
                              hipStream_t stream)
{
  (void)in_sizes; (void)n_in; (void)out_size; (void)ws_size;
  const float* x         = (const float*)d_in[0];
  const int*   y         = (const int*)  d_in[1];
  const float* patch_w   = (const float*)d_in[2];
  const float* patch_b   = (const float*)d_in[3];
  const float* cls_token = (const float*)d_in[4];
  const float* pos_embed = (const float*)d_in[5];
  const float* wqkv      = (const float*)d_in[6];
  const float* bqkv      = (const float*)d_in[7];
  const float* wo        = (const float*)d_in[8];
  const float* bo        = (const float*)d_in[9];
  const float* ln1_g     = (const float*)d_in[10];
  const float* ln1_b     = (const float*)d_in[11];
  const float* ln2_g     = (const float*)d_in[12];
  const float* ln2_b     = (const float*)d_in[13];
  const float* dw1       = (const float*)d_in[14];
  const float* db1       = (const float*)d_in[15];
  const float* dw2       = (const float*)d_in[16];
  const float* db2       = (const float*)d_in[17];
  const float* gate_w    = (const float*)d_in[18];
  const float* ew1       = (const float*)d_in[19];
  const float* eb1       = (const float*)d_in[20];
  const float* ew2       = (const float*)d_in[21];
  const float* eb2       = (const float*)d_in[22];
  const float* dec_w     = (const float*)d_in[23];
  const float* dec_b     = (const float*)d_in[24];

  const int S = NTOK, D = EMSIZE, C = CAPC, E = NEXPERT;

  // ---- workspace layout (floats) ----
  float* ws = (float*)d_ws;
  size_t off = 0;
  float* h     = ws + off; off += (size_t)S * D;          // activations
  float* bufA  = ws + off; off += (size_t)E * C * NHID;   // qkv / ffn hidden / expert hidden (max)
  float* bufB  = ws + off; off += (size_t)E * C * D;      // attn heads / ffn out / moe out / dec logits
  float* bufC  = ws + off; off += (size_t)E * C * D;      // proj out / expert oe
  float* glog  = ws + off; off += (size_t)S * E;
  float* g1    = ws + off; off += S;
  float* g2    = ws + off; off += S;
  float* gate1 = ws + off; off += S;
  float* gate2 = ws + off; off += S;
  float* lossb = ws + off; off += 256;
  int* ib = (int*)(ws + off);
  int* e1v = ib;            ib += S;
  int* e2v = ib;            ib += S;
  int* p1  = ib;            ib += S;
  int* p2  = ib;            ib += S;
  int* p1c = ib;            ib += S;
  int* p2c = ib;            ib += S;
  int* slot = ib;           ib += E * C;

  // ---- patch embed + cls + pos ----
  {
    long total = (long)BATCHN * 64 * 48;
    patch_rearrange_kernel<<<(total + 255) / 256, 256, 0, stream>>>(x, bufA);
  }
  launch_gemm(bufA, patch_w, patch_b, bufB, BATCHN * 64, D, 48, 48, D, D, nullptr, 0, stream);
  {
    long total = (long)S * D;
    assemble_kernel<<<(total + 255) / 256, 256, 0, stream>>>(bufB, cls_token, pos_embed, h);
  }

  int di = 0, mi = 0;
  for (int i = 0; i < NLAYERS; ++i) {
    // ---- attention ----
    launch_gemm(h, wqkv + (size_t)i * D * 3 * D, bqkv + (size_t)i * 3 * D, bufA,
                S, 3 * D, D, D, 3 * D, 3 * D, nullptr, 0, stream);
    attn_kernel<<<BATCHN * NHEADS, 128, 0, stream>>>(bufA, bufB);
    launch_gemm(bufB, wo + (size_t)i * D * D, bo + (size_t)i * D, bufC,
                S, D, D, D, D, D, nullptr, 0, stream);
    add_ln_kernel<<<S, 256, 0, stream>>>(h, bufC, ln1_g + (size_t)i * D, ln1_b + (size_t)i * D, h);

    if ((i & 1) == 0) {
      // ---- dense FFN ----
      launch_gemm(h, dw1 + (size_t)di * D * NHID, db1 + (size_t)di * NHID, bufA,
                  S, NHID, D, D, NHID, NHID, nullptr, 1, stream);
      launch_gemm(bufA, dw2 + (size_t)di * NHID * D, db2 + (size_t)di * D, bufB,
                  S, D, NHID, NHID, D, D, nullptr, 0, stream);
      ++di;
    } else {
      // ---- MoE FFN (GShard top-2, capacity 2600) ----
      gate_logits_kernel<<<(S + 7) / 8, 256, 0, stream>>>(h, gate_w + (size_t)mi * D * E, glog, S);
      gate_top2_kernel<<<(S + 255) / 256, 256, 0, stream>>>(glog, e1v, e2v, gate1, gate2, S);
      slot_init_kernel<<<(E * C + 255) / 256, 256, 0, stream>>>(slot, E * C);
      moe_pos_kernel<<<1, 32, 0, stream>>>(e1v, e2v, p1, p2, S);
      moe_finalize_kernel<<<(S + 255) / 256, 256, 0, stream>>>(gate1, gate2, e1v, e2v, p1, p2,
                                                               g1, g2, p1c, p2c, slot, S, C);
      for (int e = 0; e < E; ++e) {
        const float* W1 = ew1 + ((size_t)mi * E + e) * D * NHID;
        const float* B1 = eb1 + ((size_t)mi * E + e) * NHID;
        const float* W2 = ew2 + ((size_t)mi * E + e) * NHID * D;
        const float* B2 = eb2 + ((size_t)mi * E + e) * D;
        launch_gemm(h, W1, B1, bufA + (size_t)e * C * NHID,
                    C, NHID, D, D, NHID, NHID, slot + e * C, 1, stream);
        launch_gemm(bufA + (size_t)e * C * NHID, W2, B2, bufC + (size_t)e * C * D,
                    C, D, NHID, NHID, D, D, nullptr, 0, stream);
      }
      {
        long total = (long)S * (D / 4);
        moe_combine_kernel<<<(total + 255) / 256, 256, 0, stream>>>(bufC, e1v, e2v, p1c, p2c,
                                                                    g1, g2, bufB, S, C);
      }
      ++mi;
    }
    add_ln_kernel<<<S, 256, 0, stream>>>(h, bufB, ln2_g + (size_t)i * D, ln2_b + (size_t)i * D, h);
  }

  // ---- decoder on CLS tokens (row stride SEQLEN*D gathers h[:,0]) ----
  launch_gemm(h, dec_w, dec_b, bufB, BATCHN, NCLASSES, D, SEQLEN * D, NCLASSES, NCLASSES,
              nullptr, 0, stream);
  loss_per_b_kernel<<<BATCHN, 256, 0, stream>>>(bufB, y, lossb);
  loss_reduce_kernel<<<1, 256, 0, stream>>>(lossb, (float*)d_out);
}